// SAGPool_79345225826537
// MI455X (gfx1250) — compile-verified
//
#include <hip/hip_runtime.h>
#include <hip/hip_bf16.h>
#include <math.h>

#define FEAT 128

typedef __attribute__((ext_vector_type(2))) float v2f;
typedef __attribute__((ext_vector_type(8))) float v8f;
typedef __attribute__((ext_vector_type(4))) unsigned int v4u;
typedef __attribute__((ext_vector_type(8))) int v8i_t;
typedef __attribute__((ext_vector_type(4))) int v4i_t;

__device__ __forceinline__ unsigned f2key(float f) {
  unsigned u = __float_as_uint(f);
  // order-preserving flip: larger float -> larger unsigned key
  return (u & 0x80000000u) ? ~u : (u | 0x80000000u);
}

// ---------------------------------------------------------------------------
// Kernel 1: dual GEMV via WMMA f32 16x16x4.  One wave = 16 vertex rows.
// A (16x4 f32): row 0 = w0_w chunk, row 1 = w1_w chunk, rows 2..15 = 0.
// B (4x16 f32): B[k][n] = verts[row0+n][kchunk+k]  (vertex tile from LDS,
//               staged by the Tensor Data Mover).
// D row 0 = w0-dots of 16 vertices, D row 1 = w1-dots.
// Per-lane operand addressing is branch-free: one pointer select pre-loop.
// ---------------------------------------------------------------------------
__global__ __launch_bounds__(128) void k_attn_gemv(
    const float* __restrict__ verts,
    const float* __restrict__ w0w, const float* __restrict__ w0b,
    const float* __restrict__ w1w, const float* __restrict__ w1b,
    float* __restrict__ attn, float* __restrict__ vw1, int V)
{
  __shared__ float smem[4 * 16 * FEAT + 3 * FEAT];
  float* w0s  = smem + 4 * 16 * FEAT;   // 128
  float* w1s  = w0s + FEAT;             // 128
  float* zbuf = w1s + FEAT;             // 128 zeros (A rows 2..15)
  const int t = threadIdx.x;
  if (t < FEAT) { w0s[t] = w0w[t]; w1s[t] = w1w[t]; zbuf[t] = 0.0f; }
  __syncthreads();

  const int waveIn   = t >> 5;
  const int lane     = t & 31;
  const int numTiles = (V + 15) >> 4;
  const int tile     = blockIdx.x * 4 + waveIn;
  if (tile >= numTiles) return;            // wave-uniform exit, EXEC stays all-1s

  float* myTile = smem + waveIn * (16 * FEAT);
  const int row0 = tile << 4;
  const int rows = min(16, V - row0);

#if __has_builtin(__builtin_amdgcn_tensor_load_to_lds) && __has_builtin(__builtin_amdgcn_s_wait_tensorcnt)
  {
    // --- Tensor Data Mover descriptor (D#), groups per CDNA5 ISA
    unsigned long long ga = (unsigned long long)(const void*)(verts + (size_t)row0 * FEAT);
    unsigned ga_lo = (unsigned)__builtin_amdgcn_readfirstlane((int)(ga & 0xFFFFFFFFull));
    unsigned ga_hi = (unsigned)__builtin_amdgcn_readfirstlane((int)(ga >> 32));
    // low 32 bits of a generic LDS pointer are the LDS byte address
    unsigned lds   = (unsigned)__builtin_amdgcn_readfirstlane((int)(unsigned long long)(void*)myTile);
    unsigned urows = (unsigned)__builtin_amdgcn_readfirstlane(rows);

    v4u g0;
    g0.x = 1u;                                   // count=1 valid user descriptor
    g0.y = lds;                                  // lds_addr (bytes)
    g0.z = ga_lo;                                // global_addr[31:0]
    g0.w = (ga_hi & 0x01FFFFFFu) | (2u << 30);   // global_addr[56:32] | type=2 (image)

    v8i_t g1;
    g1[0] = (int)(2u << 16);                     // data_size = 4 bytes
    g1[1] = (int)(128u << 16);                   // tensor_dim0 = 128
    g1[2] = (int)(urows << 16);                  // tensor_dim1 = rows
    g1[3] = (int)(128u << 16);                   // tile_dim0 = 128
    g1[4] = (int)urows;                          // tile_dim1 = rows, tile_dim2 = 0
    g1[5] = 128;                                 // tensor_dim0_stride = 128
    g1[6] = 0;
    g1[7] = 0;
    v4i_t gz;  gz[0] = 0; gz[1] = 0; gz[2] = 0; gz[3] = 0;
    v8i_t g4z; g4z[0] = 0; g4z[1] = 0; g4z[2] = 0; g4z[3] = 0;
    g4z[4] = 0; g4z[5] = 0; g4z[6] = 0; g4z[7] = 0;

    // 6-arg form (clang-23 / therock headers): extra int32x8 SGPR group, zeroed
    __builtin_amdgcn_tensor_load_to_lds(g0, g1, gz, gz, g4z, 0);
    __builtin_amdgcn_s_wait_tensorcnt(0);
  }
#else
  for (int i = lane; i < rows * FEAT; i += 32)
    myTile[i] = verts[(size_t)row0 * FEAT + i];
#endif

  // --- branch-free per-lane operand pointers (selected once) ---
  // A layout (16x4 f32): lane L holds row M=L&15; VGPR0/1 = K = (L<16 ? 0,1 : 2,3)
  // B layout (4x16 f32): lane L holds col N=L&15; VGPR0/1 = K = (L<16 ? 0,1 : 2,3)
  const int m    = lane & 15;
  const int half = lane >> 4;
  const float* awp  = (m == 0) ? w0s : (m == 1) ? w1s : zbuf;  // A: weight rows
  const float* brow = myTile + m * FEAT;                       // B: vertex row

  v8f c = {};
#pragma unroll
  for (int kc = 0; kc < 32; ++kc) {
    const int k0 = kc * 4 + half * 2;
    v2f a = *reinterpret_cast<const v2f*>(awp + k0);
    v2f b = *reinterpret_cast<const v2f*>(brow + k0);
    c = __builtin_amdgcn_wmma_f32_16x16x4_f32(false, a, false, b, (short)0, c, false, false);
  }

  // D layout: VGPR r, lanes 0-15 -> M=r, N=lane.  Row 0 = w0 dots, row 1 = w1 dots.
  if (lane < 16 && lane < rows) {
    attn[row0 + lane] = c[0] + w0b[0];
    vw1 [row0 + lane] = c[1] + w1b[0];
  }
}

// ---------------------------------------------------------------------------
// Kernel 2: edge scatter-add  attn[r] += vw1[c]; attn[c] += vw1[r]
// ---------------------------------------------------------------------------
__global__ void k_edge_scatter(const int* __restrict__ edges,
                               const float* __restrict__ vw1,
                               float* __restrict__ attn, int E)
{
  int e = blockIdx.x * blockDim.x + threadIdx.x;
  if (e >= E) return;
  int r = edges[2 * e], c = edges[2 * e + 1];
  atomicAdd(&attn[r], vw1[c]);
  atomicAdd(&attn[c], vw1[r]);
}

// ---------------------------------------------------------------------------
// Radix-select (MSB-first, 4 rounds of 8 bits) for exact k-th largest key.
// ---------------------------------------------------------------------------
__global__ void k_init_state(unsigned* state, unsigned* bins, int k)
{
  int t = blockIdx.x * blockDim.x + threadIdx.x;
  if (t < 256) bins[t] = 0u;
  if (t == 256) { state[0] = 0u; state[1] = (unsigned)k; }
}

__global__ void k_hist(const float* __restrict__ attn, unsigned* __restrict__ bins,
                       const unsigned* __restrict__ state, int V, int shift)
{
  __shared__ unsigned h[256];
  if (threadIdx.x < 256) h[threadIdx.x] = 0u;
  __syncthreads();
  const unsigned prefix = state[0];
  const unsigned hmask  = (shift == 24) ? 0u : (0xFFFFFFFFu << (shift + 8));
  for (int v = blockIdx.x * blockDim.x + threadIdx.x; v < V; v += gridDim.x * blockDim.x) {
    unsigned key = f2key(attn[v]);
    if ((key & hmask) == (prefix & hmask))
      atomicAdd(&h[(key >> shift) & 0xFFu], 1u);
  }
  __syncthreads();
  if (threadIdx.x < 256 && h[threadIdx.x]) atomicAdd(&bins[threadIdx.x], h[threadIdx.x]);
}

__global__ void k_pick(unsigned* bins, unsigned* state, int shift)
{
  if (threadIdx.x != 0 || blockIdx.x != 0) return;
  unsigned needed = state[1];
  unsigned acc = 0;
  int b = 255;
  for (; b > 0; --b) {
    unsigned cnt = bins[b];
    if (acc + cnt >= needed) break;
    acc += cnt;
  }
  state[0] |= ((unsigned)b << shift);
  state[1]  = needed - acc;
  for (int i = 0; i < 256; ++i) bins[i] = 0u;
}

// ---------------------------------------------------------------------------
// Dual prefix scan over V (single block): mask[v] = new index (asc order) or -1.
// sel(v) = key>T  ||  (key==T && tieRankExcl(v) < needed)   [JAX tie semantics]
// newIdx(v) = #greater before v + min(tieRankExcl(v), needed)
// ---------------------------------------------------------------------------
__global__ __launch_bounds__(1024) void k_mask_scan(
    const float* __restrict__ attn, const unsigned* __restrict__ state,
    int* __restrict__ mask, int V)
{
  __shared__ unsigned sg[1024];
  __shared__ unsigned st[1024];
  const unsigned T = state[0];
  const unsigned needed = state[1];
  const int tid = threadIdx.x;
  const int chunk = (V + 1023) / 1024;
  const int lo = tid * chunk;
  const int hi = min(lo + chunk, V);

  unsigned g = 0, ties = 0;
  for (int v = lo; v < hi; ++v) {
    unsigned key = f2key(attn[v]);
    g    += (key > T) ? 1u : 0u;
    ties += (key == T) ? 1u : 0u;
  }
  sg[tid] = g; st[tid] = ties;
  __syncthreads();
  for (int off = 1; off < 1024; off <<= 1) {
    unsigned ag = (tid >= off) ? sg[tid - off] : 0u;
    unsigned at = (tid >= off) ? st[tid - off] : 0u;
    __syncthreads();
    sg[tid] += ag; st[tid] += at;
    __syncthreads();
  }
  unsigned gEx = tid ? sg[tid - 1] : 0u;
  unsigned tEx = tid ? st[tid - 1] : 0u;
  for (int v = lo; v < hi; ++v) {
    unsigned key = f2key(attn[v]);
    bool isG = (key > T);
    bool isT = (key == T);
    bool sel = isG || (isT && tEx < needed);
    int idx = (int)(gEx + (tEx < needed ? tEx : needed));
    mask[v] = sel ? idx : -1;
    gEx += isG ? 1u : 0u;
    tEx += isT ? 1u : 0u;
  }
}

// ---------------------------------------------------------------------------
// verts_up[j] = verts[v] * tanh(attn[v]) for selected v (j = mask[v])
// ---------------------------------------------------------------------------
__global__ __launch_bounds__(128) void k_gather_scale(
    const float* __restrict__ verts, const float* __restrict__ attn,
    const int* __restrict__ mask, float* __restrict__ out0, int V)
{
  int v = blockIdx.x;
  if (v >= V) return;
  int j = mask[v];
  if (j < 0) return;
  float s = tanhf(attn[v]);
  for (int f = threadIdx.x; f < FEAT; f += blockDim.x)
    out0[(size_t)j * FEAT + f] = verts[(size_t)v * FEAT + f] * s;
}

__global__ void k_vidx(float* out2, int k)
{
  int i = blockIdx.x * blockDim.x + threadIdx.x;
  if (i < k) out2[i] = (float)i;
}

// ---------------------------------------------------------------------------
// Edge compaction: stable partition (valid first, original order preserved)
// ---------------------------------------------------------------------------
__global__ __launch_bounds__(256) void k_edge_partial(
    const int* __restrict__ edges, const int* __restrict__ mask,
    unsigned* __restrict__ bsums, int E)
{
  __shared__ unsigned s[256];
  int e = blockIdx.x * 256 + threadIdx.x;
  unsigned v = 0;
  if (e < E) {
    int r = edges[2 * e], c = edges[2 * e + 1];
    v = (mask[r] >= 0 && mask[c] >= 0) ? 1u : 0u;
  }
  s[threadIdx.x] = v;
  __syncthreads();
  for (int off = 128; off > 0; off >>= 1) {
    if (threadIdx.x < off) s[threadIdx.x] += s[threadIdx.x + off];
    __syncthreads();
  }
  if (threadIdx.x == 0) bsums[blockIdx.x] = s[0];
}

__global__ __launch_bounds__(1024) void k_scan_bsums(
    unsigned* __restrict__ bsums, int nb,
    float* __restrict__ out_nedges, unsigned* __restrict__ nedges_ws)
{
  __shared__ unsigned ss[1024];
  const int tid = threadIdx.x;
  const int chunk = (nb + 1023) / 1024;
  const int lo = tid * chunk;
  const int hi = min(lo + chunk, nb);
  unsigned t = 0;
  for (int i = lo; i < hi; ++i) t += bsums[i];
  ss[tid] = t;
  __syncthreads();
  for (int off = 1; off < 1024; off <<= 1) {
    unsigned a = (tid >= off) ? ss[tid - off] : 0u;
    __syncthreads();
    ss[tid] += a;
    __syncthreads();
  }
  unsigned excl = tid ? ss[tid - 1] : 0u;
  for (int i = lo; i < hi; ++i) { unsigned b = bsums[i]; bsums[i] = excl; excl += b; }
  if (tid == 1023) { *nedges_ws = ss[1023]; *out_nedges = (float)ss[1023]; }
}

__global__ __launch_bounds__(256) void k_edge_scatter_out(
    const int* __restrict__ edges, const int* __restrict__ mask,
    const unsigned* __restrict__ bsums, const unsigned* __restrict__ nedges_ws,
    float* __restrict__ out1, float* __restrict__ out3, int E)
{
  __shared__ unsigned s[256];
  int e = blockIdx.x * 256 + threadIdx.x;
  int mr = -1, mc = -1;
  unsigned valid = 0;
  if (e < E) {
    int r = edges[2 * e], c = edges[2 * e + 1];
    mr = mask[r]; mc = mask[c];
    valid = (mr >= 0 && mc >= 0) ? 1u : 0u;
  }
  s[threadIdx.x] = valid;
  __syncthreads();
  for (int off = 1; off < 256; off <<= 1) {
    unsigned a = (threadIdx.x >= off) ? s[threadIdx.x - off] : 0u;
    __syncthreads();
    s[threadIdx.x] += a;
    __syncthreads();
  }
  if (e < E) {
    unsigned excl = s[threadIdx.x] - valid;     // valid count before me in block
    unsigned validBefore = bsums[blockIdx.x] + excl;
    if (valid) {
      size_t pos = validBefore;
      out1[2 * pos]     = (float)mr;
      out1[2 * pos + 1] = (float)mc;
      out3[pos]         = (float)e;
    } else {
      size_t pos = (size_t)(*nedges_ws) + (unsigned)e - validBefore;
      out1[2 * pos]     = -1.0f;
      out1[2 * pos + 1] = -1.0f;
      out3[pos]         = -1.0f;
    }
  }
}

// ---------------------------------------------------------------------------
extern "C" void kernel_launch(void* const* d_in, const int* in_sizes, int n_in,
                              void* d_out, int out_size, void* d_ws, size_t ws_size,
                              hipStream_t stream)
{
  const float* verts = (const float*)d_in[0];
  const int*   edges = (const int*)d_in[1];
  // d_in[2] = verts_idx (arange), d_in[3] = edges_idx (arange): values implied
  const float* w0w = (const float*)d_in[4];
  const float* w0b = (const float*)d_in[5];
  const float* w1w = (const float*)d_in[6];
  const float* w1b = (const float*)d_in[7];

  const int V = in_sizes[0] / FEAT;
  const int E = in_sizes[1] / 2;
  const int k = (int)(((long long)V * 4) / 5);   // RATIO = 0.8

  // output regions (all float per harness convention)
  float* out0 = (float*)d_out;                   // verts_up      k*FEAT
  float* out1 = out0 + (size_t)k * FEAT;         // new_edges     E*2
  float* out2 = out1 + (size_t)E * 2;            // verts_idx_up  k
  float* out3 = out2 + (size_t)k;                // edges_idx_up  E
  float* out4 = out3 + (size_t)E;                // n_edges       1

  // workspace layout
  float*    attn      = (float*)d_ws;            // V
  float*    vw1       = attn + V;                // V
  int*      mask      = (int*)(vw1 + V);         // V
  unsigned* bins      = (unsigned*)(mask + V);   // 256
  unsigned* state     = bins + 256;              // 8
  unsigned* nedges_ws = state + 8;               // 8 (aligned slot)
  unsigned* bsums     = nedges_ws + 8;           // ceil(E/256)

  const int numTiles = (V + 15) / 16;
  const int gemvBlocks = (numTiles + 3) / 4;
  const int nbE = (E + 255) / 256;

  k_attn_gemv<<<gemvBlocks, 128, 0, stream>>>(verts, w0w, w0b, w1w, w1b, attn, vw1, V);
  k_edge_scatter<<<(E + 255) / 256, 256, 0, stream>>>(edges, vw1, attn, E);

  k_init_state<<<1, 512, 0, stream>>>(state, bins, k);
  for (int shift = 24; shift >= 0; shift -= 8) {
    k_hist<<<512, 256, 0, stream>>>(attn, bins, state, V, shift);
    k_pick<<<1, 1, 0, stream>>>(bins, state, shift);
  }

  k_mask_scan<<<1, 1024, 0, stream>>>(attn, state, mask, V);

  k_gather_scale<<<V, 128, 0, stream>>>(verts, attn, mask, out0, V);
  k_vidx<<<(k + 255) / 256, 256, 0, stream>>>(out2, k);

  k_edge_partial<<<nbE, 256, 0, stream>>>(edges, mask, bsums, E);
  k_scan_bsums<<<1, 1024, 0, stream>>>(bsums, nbE, out4, nedges_ws);
  k_edge_scatter_out<<<nbE, 256, 0, stream>>>(edges, mask, bsums, nedges_ws, out1, out3, E);
}